// GF_23502061044320
// MI455X (gfx1250) — compile-verified
//
#include <hip/hip_runtime.h>
#include <stdint.h>

#define KR   5            // kernel radius
#define TS   64           // output tile edge
#define PT   74           // padded tile edge = TS + 2*KR
#define LSTR 77           // LDS row stride in floats (bank-conflict friendly)
#define IMG  512

// reflect-pad index mapping for padded coord i in [-5, IMG+4]:
//   i < 0      -> -i
//   i >= IMG   -> 2*IMG-2-i
__device__ __forceinline__ unsigned refl(int i) {
  i = i < 0 ? -i : i;
  int m = 2 * IMG - 2 - i;          // 1022 - i
  return (unsigned)(i < m ? i : m); // min handles the top edge
}

__global__ __launch_bounds__(256)
void GF_boxlog_kernel(const float* __restrict__ x, float* __restrict__ out) {
  __shared__ float tile[PT * LSTR]; // raw (unshifted) padded input tile
  __shared__ float s6a [TS * LSTR]; // 6-row top sums   (shift folded: +0.06)
  __shared__ float s6b [TS * LSTR]; // 6-row bottom sums (shift folded: +0.06)

  const int tid = threadIdx.x;
  const int tx0 = blockIdx.x * TS;
  const int ty0 = blockIdx.y * TS;
  const int b   = blockIdx.z;

  // ---------------- Phase 0: async reflect-gather into LDS ----------------
  // CDNA5 path: global_load_async_to_lds_b32 (GVS mode), tracked by ASYNCcnt.
  const unsigned tileBase = (unsigned)(uintptr_t)(&tile[0]); // low 32 bits == LDS offset
  const unsigned batchOff = (unsigned)b * (unsigned)(IMG * IMG * 4);
  for (int idx = tid; idx < PT * PT; idx += 256) {
    int row = idx / PT;
    int col = idx - row * PT;
    unsigned sy   = refl(ty0 + row - KR);
    unsigned sx   = refl(tx0 + col - KR);
    unsigned voff = batchOff + (sy * (unsigned)IMG + sx) * 4u;
    unsigned lofs = tileBase + (unsigned)(row * LSTR + col) * 4u;
    asm volatile("global_load_async_to_lds_b32 %0, %1, %2"
                 :: "v"(lofs), "v"(voff), "s"(x) : "memory");
  }
  asm volatile("s_wait_asynccnt 0" ::: "memory");
  __syncthreads();

  // ------------- Phase 1: vertical 6-row sliding sums (s6a/s6b) -------------
  // Units: 74 columns x 4 row-segments of 16 = 296 units.
  for (int u = tid; u < PT * 4; u += 256) {
    int xc = u % PT;
    int ys = (u / PT) * 16;
    float a  = 0.06f; // 6 * 0.01 shift folded in
    float bb = 0.06f;
#pragma unroll
    for (int r = 0; r < 6; ++r)  a  += tile[(ys + r) * LSTR + xc];
#pragma unroll
    for (int r = 5; r < 11; ++r) bb += tile[(ys + r) * LSTR + xc];
    s6a[ys * LSTR + xc] = a;
    s6b[ys * LSTR + xc] = bb;
#pragma unroll
    for (int y = ys + 1; y < ys + 16; ++y) {
      a  += tile[(y + 5)  * LSTR + xc] - tile[(y - 1) * LSTR + xc];
      bb += tile[(y + 10) * LSTR + xc] - tile[(y + 4) * LSTR + xc];
      s6a[y * LSTR + xc] = a;
      s6b[y * LSTR + xc] = bb;
    }
  }
  __syncthreads();

  // --------- Phase 2: horizontal sliding sums + log/sqrt + store ---------
  const int oy = tid >> 2;               // output row within tile (0..63)
  const int x0 = (tid & 3) * 16;         // 16-px segment start (0,16,32,48)
  const float* ta = &s6a[oy * LSTR];
  const float* tb = &s6b[oy * LSTR];
  const float* tm = &tile[(oy + 5) * LSTR]; // middle row of the 11-row window

  // s11[c] = 11-row column sum at padded col c (with +0.01 shift folded)
  auto s11v = [&](int c) -> float { return ta[c] + tb[c] - tm[c] - 0.01f; };

  float gx1 = 0.f, gx2 = 0.f, gy1 = 0.f, gy2 = 0.f;
#pragma unroll
  for (int c = 0; c < 6; ++c)  gx1 += s11v(x0 + c);        // cols x..x+5
#pragma unroll
  for (int c = 5; c < 11; ++c) gx2 += s11v(x0 + c);        // cols x+5..x+10
#pragma unroll
  for (int c = 0; c < 11; ++c) { gy1 += ta[x0 + c]; gy2 += tb[x0 + c]; }

  const float LN2 = 0.6931471805599453f;
  float* orow = out + (((size_t)b * IMG + (size_t)(ty0 + oy)) * IMG) + tx0 + x0;
  float4 v;
#pragma unroll
  for (int px = 0; px < 16; ++px) {
    float gx = (__builtin_amdgcn_logf(gx1) - __builtin_amdgcn_logf(gx2)) * LN2;
    float gy = (__builtin_amdgcn_logf(gy1) - __builtin_amdgcn_logf(gy2)) * LN2;
    float r  = __builtin_amdgcn_sqrtf(gx * gx + gy * gy);
    (&v.x)[px & 3] = r;
    if ((px & 3) == 3) ((float4*)orow)[px >> 2] = v;  // global_store_b128
    if (px < 15) {
      int xx = x0 + px;
      gx1 += s11v(xx + 6)  - s11v(xx);
      gx2 += s11v(xx + 11) - s11v(xx + 5);
      gy1 += ta[xx + 11] - ta[xx];
      gy2 += tb[xx + 11] - tb[xx];
    }
  }
}

extern "C" void kernel_launch(void* const* d_in, const int* in_sizes, int n_in,
                              void* d_out, int out_size, void* d_ws, size_t ws_size,
                              hipStream_t stream) {
  const float* x = (const float*)d_in[0];
  float* out     = (float*)d_out;
  int B = in_sizes[0] / (IMG * IMG);   // 64
  dim3 grid(IMG / TS, IMG / TS, B);    // 8 x 8 x 64 tiles
  GF_boxlog_kernel<<<grid, 256, 0, stream>>>(x, out);
}